// Condensate_17257178595436
// MI455X (gfx1250) — compile-verified
//
#include <hip/hip_runtime.h>
#include <stdint.h>

typedef __attribute__((ext_vector_type(2))) float v2f;
typedef __attribute__((ext_vector_type(8))) float v8f;

#define R2 0.49f            // T_D^2
#define KEY_TB 0xBE4CCCCDu  // order-key of T_B = 0.2f

// order-preserving uint key for floats (total order, works for negatives)
__device__ __forceinline__ uint32_t okey(float b) {
    uint32_t u = __float_as_uint(b);
    u ^= (uint32_t)((int32_t)u >> 31) | 0x80000000u;
    return u;
}

__global__ __launch_bounds__(1024)
void condensate_kernel(const float* __restrict__ x, uint8_t* __restrict__ iscond,
                       int* __restrict__ ncond, int P) {
    const int e    = blockIdx.x;
    const size_t ebase = (size_t)e * (size_t)P;
    const int tid  = threadIdx.x;
    const int lane = tid & 31;
    const int wave = tid >> 5;
    const int half = lane >> 4;
    const int l16  = lane & 15;
    const int ngr  = P >> 8;        // 256-point groups
    const int gpw  = ngr >> 5;      // groups per wave (8 for P=65536)

    __shared__ uint32_t abw[2048];              // assigned bitmap, P/32 words
    __shared__ unsigned long long grp[256];     // per-group max key (okey(beta)<<32 | ~idx)
    __shared__ float rx[16], ry[16], rz[16];    // accepted refs this batch
    __shared__ float alds0[32], alds1[32];      // preformatted WMMA A operand (refs)
    __shared__ float rn_sh[16];                 // |r|^2 per ref slot
    __shared__ int nb_sh, done_sh, cnt_sh;

    // ---- init: clear assigned bitmap, clear iscond slice, counters ----
    for (int i = tid; i < (P >> 5); i += 1024) abw[i] = 0u;
    {
        int* ic = (int*)(iscond + ebase);
        for (int i = tid; i < (P >> 2); i += 1024) ic[i] = 0;
    }
    if (tid == 0) cnt_sh = 0;
    __syncthreads();

    // ---- initial group maxima ----
    for (int gg = 0; gg < gpw; ++gg) {
        int g = wave * gpw + gg;
        unsigned long long rm = 0ull;
        for (int j = lane; j < 256; j += 32) {
            int p = (g << 8) + j;
            float bt = x[(ebase + p) * 17 + 9];
            unsigned long long key =
                ((unsigned long long)okey(bt) << 32) | (0xFFFFFFFFu - (uint32_t)p);
            rm = rm > key ? rm : key;
        }
#pragma unroll
        for (int off = 16; off; off >>= 1) {
            unsigned long long o = __shfl_xor(rm, off, 32);
            rm = rm > o ? rm : o;
        }
        if (lane == 0) grp[g] = rm;
    }
    __syncthreads();

    // ---- batched greedy loop ----
    for (int iter = 0; iter < 8192; ++iter) {
        if (wave == 0) {
            int nb = 0, done = 0;
            for (int k = 0; k < 16; ++k) {
                // global argmax over group keys
                unsigned long long kb = 0ull;
                for (int j = lane; j < ngr; j += 32) {
                    unsigned long long v = grp[j];
                    kb = kb > v ? kb : v;
                }
#pragma unroll
                for (int off = 16; off; off >>= 1) {
                    unsigned long long o = __shfl_xor(kb, off, 32);
                    kb = kb > o ? kb : o;
                }
                if ((uint32_t)(kb >> 32) < KEY_TB) { done = 1; break; }
                uint32_t idx = 0xFFFFFFFFu - (uint32_t)kb;
                const float* wp = x + (ebase + idx) * 17;
                float wxx = wp[14], wyy = wp[15], wzz = wp[16];
                // reject if within radius of an earlier accepted ref of this batch
                bool nl = false;
                if (lane < nb) {
                    float dx = wxx - rx[lane], dy = wyy - ry[lane], dz = wzz - rz[lane];
                    nl = (dx * dx + dy * dy + dz * dz) <= R2;
                }
                uint32_t rj = __builtin_amdgcn_ballot_w32(nl);
                if (lane == 0) {
                    abw[idx >> 5] |= 1u << (idx & 31);
                    if (!rj) {
                        rx[nb] = wxx; ry[nb] = wyy; rz[nb] = wzz;
                        iscond[ebase + idx] = 1;
                        cnt_sh = cnt_sh + 1;
                    }
                }
                if (!rj) nb++;
                asm volatile("" ::: "memory");
                // refresh winner's group max
                int g = (int)(idx >> 8);
                unsigned long long rm = 0ull;
                for (int j = lane; j < 256; j += 32) {
                    int p = (g << 8) + j;
                    uint32_t aw = abw[p >> 5];
                    float bt = x[(ebase + p) * 17 + 9];
                    unsigned long long key = ((aw >> (p & 31)) & 1u)
                        ? 0ull
                        : (((unsigned long long)okey(bt) << 32) | (0xFFFFFFFFu - (uint32_t)p));
                    rm = rm > key ? rm : key;
                }
#pragma unroll
                for (int off = 16; off; off >>= 1) {
                    unsigned long long o = __shfl_xor(rm, off, 32);
                    rm = rm > o ? rm : o;
                }
                if (lane == 0) grp[g] = rm;
                asm volatile("" ::: "memory");
            }
            // Build WMMA A operand (refs on M): A[r] = (1, x_r, y_r, z_r)
            // layout: v0 = K0 (lanes0-15) / K2 (lanes16-31); v1 = K1 / K3
            float cxr = (l16 < nb) ? rx[l16] : 3.0e3f;
            float cyr = (l16 < nb) ? ry[l16] : 3.0e3f;
            float czr = (l16 < nb) ? rz[l16] : 3.0e3f;
            float rn = cxr * cxr + cyr * cyr + czr * czr;
            alds0[lane] = half ? cyr : 1.0f;
            alds1[lane] = half ? czr : cxr;
            if (!half) rn_sh[l16] = rn;
            if (lane == 0) { nb_sh = nb; done_sh = done; }
        }
        __syncthreads();
        int nb = nb_sh, done = done_sh;

        if (nb > 0) {
            v2f av; av.x = alds0[lane]; av.y = alds1[lane];
            // C[r][p] = |r|^2 - R2 for row r = i + 8*half  ->  D = dist^2 - R2
            v8f c8;
#pragma unroll
            for (int i = 0; i < 8; ++i) c8[i] = rn_sh[i + (half << 3)] - R2;

            for (int gg = 0; gg < gpw; ++gg) {
                int g = wave * gpw + gg;
                unsigned long long rm = 0ull;
                for (int t = 0; t < 16; ++t) {
                    int pb = (g << 8) + (t << 4);
                    int p  = pb + l16;
                    const float* cp = x + (ebase + p) * 17;
                    float cx = cp[14], cy = cp[15], cz = cp[16], bt = cp[9];
                    float pn = cx * cx + cy * cy + cz * cz;
                    // B[.][p] = (|p|^2, -2x_p, -2y_p, -2z_p): v0 = K0/K2, v1 = K1/K3
                    v2f bvv; bvv.x = half ? (-2.0f * cy) : pn;
                    bvv.y = half ? (-2.0f * cz) : (-2.0f * cx);
                    // D[r][p] = dist^2(r,p) - R2   (16 refs x 16 points)
                    v8f d = __builtin_amdgcn_wmma_f32_16x16x4_f32(
                        false, av, false, bvv, (short)0, c8, false, false);
                    // point assigned by any of this half's 8 refs?  min over rows <= 0
                    float mn = d[0];
#pragma unroll
                    for (int i = 1; i < 8; ++i) mn = __builtin_fminf(mn, d[i]);
                    uint32_t m = __builtin_amdgcn_ballot_w32(mn <= 0.0f);
                    uint32_t ab16 = (m & 0xFFFFu) | (m >> 16);   // OR both ref-halves
                    int word = pb >> 5, sh = pb & 31;
                    uint32_t aw = abw[word];
                    if (lane == 0 && ab16) abw[word] = aw | (ab16 << sh);
                    bool asg = ((aw >> (sh + l16)) & 1u) || ((ab16 >> l16) & 1u);
                    unsigned long long key = asg
                        ? 0ull
                        : (((unsigned long long)okey(bt) << 32) | (0xFFFFFFFFu - (uint32_t)p));
                    rm = rm > key ? rm : key;
                }
#pragma unroll
                for (int off = 16; off; off >>= 1) {
                    unsigned long long o = __shfl_xor(rm, off, 32);
                    rm = rm > o ? rm : o;
                }
                if (lane == 0) grp[g] = rm;
            }
        }
        __syncthreads();
        if (done) break;
    }
    if (tid == 0) ncond[e] = cnt_sh;
}

__global__ void finalize_kernel(const float* __restrict__ x, const uint8_t* __restrict__ iscond,
                                const int* __restrict__ ncond, float* __restrict__ out,
                                int N, int E) {
    size_t stride = (size_t)gridDim.x * blockDim.x;
    for (size_t p = (size_t)blockIdx.x * blockDim.x + threadIdx.x; p < (size_t)N; p += stride) {
        float m = iscond[p] ? 1.0f : 0.0f;
        const float* xp = x + p * 17;
        float* op = out + p * 17;
#pragma unroll
        for (int j = 0; j < 17; ++j) op[j] = xp[j] * m;
    }
    if (blockIdx.x == 0 && threadIdx.x == 0) {
        float* rs = out + (size_t)N * 17;
        rs[0] = 0.0f;
        int c = 0;
        for (int e2 = 0; e2 < E; ++e2) { c += ncond[e2]; rs[e2 + 1] = (float)c; }
    }
}

extern "C" void kernel_launch(void* const* d_in, const int* in_sizes, int n_in,
                              void* d_out, int out_size, void* d_ws, size_t ws_size,
                              hipStream_t stream) {
    const float* x = (const float*)d_in[0];
    (void)n_in; (void)out_size; (void)ws_size;
    int E = in_sizes[1] - 1;           // 16
    int N = in_sizes[0] / 17;          // 1048576
    int P = N / E;                     // 65536
    uint8_t* iscond = (uint8_t*)d_ws;
    int* ncond = (int*)((char*)d_ws + (((size_t)N + 255) & ~(size_t)255));
    condensate_kernel<<<E, 1024, 0, stream>>>(x, iscond, ncond, P);
    finalize_kernel<<<2048, 256, 0, stream>>>(x, iscond, ncond, (float*)d_out, N, E);
}